// LengthRegulator_81226421502579
// MI455X (gfx1250) — compile-verified
//
#include <hip/hip_runtime.h>

// Problem constants from the reference setup.
#define BB 32
#define TT 512
#define HH 512
#define H4 (HH / 4)          // 128 float4 per row
#define WAVES_PER_BLOCK 8    // 256 threads, wave32
#define FRAMES_PER_WAVE 2
#define FRAMES_PER_BLOCK (WAVES_PER_BLOCK * FRAMES_PER_WAVE)  // 16

typedef float v4f __attribute__((ext_vector_type(4)));

// ---------------------------------------------------------------------------
// Kernel 1: per-batch inclusive scan of durations (Hillis-Steele in LDS).
// One 512-thread block per batch row. Tiny (32x512 ints), runs once.
// ---------------------------------------------------------------------------
__global__ __launch_bounds__(TT) void lr_scan(const int* __restrict__ dur,
                                              int* __restrict__ cum) {
  const int b = blockIdx.x;
  const int t = threadIdx.x;
  __shared__ int s[TT];
  s[t] = dur[b * TT + t];
  __syncthreads();
#pragma unroll
  for (int off = 1; off < TT; off <<= 1) {
    int v = (t >= off) ? s[t - off] : 0;
    __syncthreads();
    s[t] += v;
    __syncthreads();
  }
  cum[b * TT + t] = s[t];
}

// ---------------------------------------------------------------------------
// Kernel 2: gather + zero-pad. Block = 256 threads handles one batch and 16
// consecutive frames (2 per wave). The cum row is staged into LDS with the
// CDNA5 async global->LDS DMA path (ASYNCcnt), searched per frame, and each
// 2 KB row is copied as 4x b128 per lane with non-temporal stores so the
// 150 MB output stream does not evict x (32 MB, L2-resident) from L2.
// ---------------------------------------------------------------------------
__global__ __launch_bounds__(256) void lr_gather(const float* __restrict__ x,
                                                 const int* __restrict__ cum,
                                                 float* __restrict__ out,
                                                 int max_len) {
  __shared__ int s_cum[TT];  // only shared object -> LDS offset 0
  const int b = blockIdx.y;
  const int tid = threadIdx.x;

  // Stage the cumsum row (2 KB) into LDS via async DMA: 128 lanes x b128.
  // LDS byte address comes from the VDST VGPR (lds_off), per 8_async_tensor.md.
  if (tid < 128) {
    unsigned lds_off = (unsigned)tid * 16u;          // s_cum starts at 0
    const int* gsrc = cum + b * TT + tid * 4;        // 16B per lane
    asm volatile("global_load_async_to_lds_b128 %0, %1, off"
                 :
                 : "v"(lds_off), "v"(gsrc)
                 : "memory");
  }
  asm volatile("s_wait_asynccnt 0x0" ::: "memory");
  __syncthreads();

  const int total = s_cum[TT - 1];
  const int wave = tid >> 5;
  const int lane = tid & 31;
  const int f0 = (blockIdx.x * WAVES_PER_BLOCK + wave) * FRAMES_PER_WAVE;

  const v4f* __restrict__ xb =
      reinterpret_cast<const v4f*>(x + (size_t)b * TT * HH);
  v4f* __restrict__ ob =
      reinterpret_cast<v4f*>(out + (size_t)b * max_len * HH);

  int tok[FRAMES_PER_WAVE];
  bool valid[FRAMES_PER_WAVE];

#pragma unroll
  for (int i = 0; i < FRAMES_PER_WAVE; ++i) {
    const int f = f0 + i;
    valid[i] = (f < max_len) && (f < total);
    // first j with cum[j] > f  (searchsorted right); all lanes identical ->
    // LDS broadcast reads, no divergence.
    int lo = 0, hi = TT;
    while (lo < hi) {
      int mid = (lo + hi) >> 1;
      if (s_cum[mid] > f) hi = mid; else lo = mid + 1;
    }
    tok[i] = (lo < TT - 1) ? lo : (TT - 1);
  }

  // Prefetch both source rows before the copy loop (global_prefetch_b8);
  // 32 lanes x 16B at v4f offsets lane and lane+64 touch the 1st and 3rd 512 B of each 2 KB row.
#pragma unroll
  for (int i = 0; i < FRAMES_PER_WAVE; ++i) {
    if (valid[i]) {
      __builtin_prefetch(xb + (size_t)tok[i] * H4 + lane, 0, 3);
      __builtin_prefetch(xb + (size_t)tok[i] * H4 + lane + 64, 0, 3);
    }
  }

#pragma unroll
  for (int i = 0; i < FRAMES_PER_WAVE; ++i) {
    const int f = f0 + i;
    if (f >= max_len) continue;  // uniform per wave
    v4f* dst = ob + (size_t)f * H4 + lane;
    if (valid[i]) {
      const v4f* src = xb + (size_t)tok[i] * H4 + lane;
      // 4 x b128 loads in flight, then 4 x b128 NT stores.
      v4f r0 = src[0];
      v4f r1 = src[32];
      v4f r2 = src[64];
      v4f r3 = src[96];
      __builtin_nontemporal_store(r0, dst);
      __builtin_nontemporal_store(r1, dst + 32);
      __builtin_nontemporal_store(r2, dst + 64);
      __builtin_nontemporal_store(r3, dst + 96);
    } else {
      v4f z = {0.f, 0.f, 0.f, 0.f};
      __builtin_nontemporal_store(z, dst);
      __builtin_nontemporal_store(z, dst + 32);
      __builtin_nontemporal_store(z, dst + 64);
      __builtin_nontemporal_store(z, dst + 96);
    }
  }
}

// ---------------------------------------------------------------------------
// Host launcher. Inputs: d_in[0]=x (f32), d_in[1]=durations (i32),
// d_in[2]=max_len (i32 scalar, unused: derived from out_size so the launch
// stays graph-capture-safe). d_ws holds cum[B,T] (64 KB).
// ---------------------------------------------------------------------------
extern "C" void kernel_launch(void* const* d_in, const int* in_sizes, int n_in,
                              void* d_out, int out_size, void* d_ws,
                              size_t ws_size, hipStream_t stream) {
  (void)in_sizes; (void)n_in; (void)ws_size;

  const float* x = (const float*)d_in[0];
  const int* dur = (const int*)d_in[1];
  float* out = (float*)d_out;
  int* cum = (int*)d_ws;  // 32*512*4 = 64 KB scratch

  const int max_len = out_size / (BB * HH);

  lr_scan<<<BB, TT, 0, stream>>>(dur, cum);

  dim3 grid((max_len + FRAMES_PER_BLOCK - 1) / FRAMES_PER_BLOCK, BB);
  lr_gather<<<grid, 256, 0, stream>>>(x, cum, out, max_len);
}